// RealSHT_2731599200378
// MI455X (gfx1250) — compile-verified
//
#include <hip/hip_runtime.h>
#include <math.h>

// CDNA5 wave32 WMMA fp32: D(16x16) = A(16x4) x B(4x16) + C
typedef __attribute__((ext_vector_type(2))) float v2f;
typedef __attribute__((ext_vector_type(8))) float v8f;
// 4-byte-aligned v2f for loads from odd-strided rows (pctw row stride = 361)
typedef v2f __attribute__((aligned(4))) v2f_a4;

#define NLAT 361
#define NLON 720
#define LMAX 360
#define MMAX 361
#define NCH  8
#define ROWS (NCH * NLAT)          // 2888  (c,k) rows for stage-1 GEMM
#define COLS (2 * MMAX)            // 722   (m, re/im) cols for stage-1 GEMM
#define WCOLS 736                  // 722 padded to 46 tiles of 16
#define RTILES ((ROWS + 15) / 16)  // 181
#define CTILES (WCOLS / 16)        // 46
#define LTILES ((LMAX + 15) / 16)  // 23
#define KPAIR  181                 // ceil(361/2): k-pairs per m in Xt
#define TWO_PI_F 6.283185307179586

// ---------------------------------------------------------------------------
// Kernel 0: DFT matrix in K-paired layout:
//   Wp[(n>>1)*WCOLS*2 + col*2 + (n&1)] = (2pi/720) * {cos,-sin}(2pi*m*n/720)
// so a lane's two consecutive-n B values are one b64 load.
// Exact integer reduction (m*n mod 720) keeps fp32 sincos accurate.
// ---------------------------------------------------------------------------
__global__ void build_dft_matrix(float* __restrict__ Wp) {
    int idx = blockIdx.x * blockDim.x + threadIdx.x;
    if (idx >= NLON * WCOLS) return;
    int n   = idx / WCOLS;
    int col = idx - n * WCOLS;
    int m   = col >> 1;
    int ri  = col & 1;
    int p   = (n * m) % NLON;                      // exact phase reduction
    float th = (float)p * (float)(TWO_PI_F / NLON);
    float s, c;
    sincosf(th, &s, &c);
    const float scale = (float)(TWO_PI_F / NLON);  // 2*pi * (1/720) forward norm
    Wp[((n >> 1) * WCOLS + col) * 2 + (n & 1)] = ri ? (-scale * s) : (scale * c);
}

// ---------------------------------------------------------------------------
// Kernel 1: rfft-as-GEMM via V_WMMA_F32_16X16X4_F32.
//   C[row=(c,k), col=(m,ri)] = sum_n x[row][n] * W[n][col]
// K=720 -> 180 clean steps, branch-free: OOB M rows are address-clamped
// (their D rows are dropped at store), padded N cols exist in Wp.
// Result scattered to Xt[m][k>>1][2c+ri][k&1] (K-paired for stage 2).
// x (8.3MB) + W (2.1MB) are L2-resident across all tiles.
// ---------------------------------------------------------------------------
__global__ void dft_wmma(const float* __restrict__ x,
                         const float* __restrict__ Wp,
                         float* __restrict__ Xt) {
    const int lane  = threadIdx.x & 31;
    const int wv    = threadIdx.x >> 5;
    const int rtile = blockIdx.x;                  // 0..180
    const int ctile = blockIdx.y * 4 + wv;         // 0..45 (one wave per tile)
    if (ctile >= CTILES) return;                   // wave-uniform exit

    const int r0   = rtile * 16;
    const int col0 = ctile * 16;
    const int lo   = lane & 15;
    const int hi   = lane >> 4;                    // K half-select

    const int arow   = r0 + lo;                    // A fragment row (M)
    const int arow_c = arow < ROWS ? arow : (ROWS - 1);  // clamp: dead rows dropped
    const int bcol   = col0 + lo;                  // B fragment col (N), < WCOLS

    const float* pA = x + (long)arow_c * NLON;     // + ka       (8B aligned)
    const float* pB = Wp + (long)bcol * 2;         // + (ka>>1)*WCOLS*2 (8B aligned)

    v8f acc = {};
#pragma unroll 4
    for (int k = 0; k < NLON; k += 4) {
        const int ka = k + hi * 2;
        v2f a = *(const v2f*)(pA + ka);
        v2f b = *(const v2f*)(pB + (long)(ka >> 1) * (WCOLS * 2));
        acc = __builtin_amdgcn_wmma_f32_16x16x4_f32(
            false, a, false, b, (short)0, acc, false, false);
    }

    if (bcol >= COLS) return;                      // padded m columns: drop
    const int m  = bcol >> 1;
    const int ri = bcol & 1;
#pragma unroll
    for (int i = 0; i < 8; ++i) {
        const int row = r0 + i + 8 * hi;           // D layout: VGPR i -> M row
        if (row < ROWS) {
            const int cch = row / NLAT;
            const int kk  = row - cch * NLAT;
            // K-paired spectrum: Xt[m][kk>>1][2c+ri][kk&1]
            Xt[(((long)m * KPAIR + (kk >> 1)) * 16 + cch * 2 + ri) * 2 + (kk & 1)] = acc[i];
        }
    }
}

// ---------------------------------------------------------------------------
// Kernel 2: per-m Legendre contraction via V_WMMA_F32_16X16X4_F32.
//   y[l, (c,ri)] = sum_k pctw[m][l][k] * Xt[m][k][(c,ri)]
// M = l (23 tiles), N = 16 = 8 channels x {re,im} (fully packed), K = 361.
// Branch-free main loop (k=0..359, b64 fragment loads) + explicit k=360 tail
// with dead K slots zeroed in registers. pctw streams once from HBM
// (187.6 MB -> the ~8us roofline); Xt[m] panels are L2-hot.
// ---------------------------------------------------------------------------
__global__ void legendre_wmma(const float* __restrict__ pctw,
                              const float* __restrict__ Xt,
                              float* __restrict__ out) {
    const int lane  = threadIdx.x & 31;
    const int wv    = threadIdx.x >> 5;
    const int m     = blockIdx.x;                  // 0..360
    const int ltile = blockIdx.y * 4 + wv;         // 0..22
    if (ltile >= LTILES) return;                   // wave-uniform exit

    const int l0 = ltile * 16;
    const int lo = lane & 15;
    const int hi = lane >> 4;

    const int al   = l0 + lo;                      // A fragment row (M) = l
    const int al_c = al < LMAX ? al : (LMAX - 1);  // clamp: dead rows dropped
    const float* pA = pctw + ((long)m * LMAX + al_c) * NLAT;      // 4B aligned
    const float* pB = Xt + ((long)m * KPAIR * 16 + lo) * 2;       // 8B aligned

    v8f acc = {};
#pragma unroll 6
    for (int k = 0; k < 360; k += 4) {             // 90 clean steps
        const int ka = k + hi * 2;
        v2f a = *(const v2f_a4*)(pA + ka);
        v2f b = *(const v2f*)(pB + (long)(ka >> 1) * 32);
        acc = __builtin_amdgcn_wmma_f32_16x16x4_f32(
            false, a, false, b, (short)0, acc, false, false);
    }
    {   // tail: k = 360 only (K slots 361..363 hard-zeroed, no OOB/poison reads)
        v2f a, b;
        a.x = (hi == 0) ? pA[360] : 0.0f;
        a.y = 0.0f;
        b.x = (hi == 0) ? pB[(360 >> 1) * 32] : 0.0f;   // pair 180, slot 0
        b.y = 0.0f;
        acc = __builtin_amdgcn_wmma_f32_16x16x4_f32(
            false, a, false, b, (short)0, acc, false, false);
    }

    // out[c][l][m] complex interleaved: flat = ((c*360 + l)*361 + m)*2 + ri
    const int cch = lo >> 1;
    const int ri  = lo & 1;
#pragma unroll
    for (int i = 0; i < 8; ++i) {
        const int l = l0 + i + 8 * hi;
        if (l < LMAX)
            out[(((long)cch * LMAX + l) * MMAX + m) * 2 + ri] = acc[i];
    }
}

// ---------------------------------------------------------------------------
extern "C" void kernel_launch(void* const* d_in, const int* in_sizes, int n_in,
                              void* d_out, int out_size, void* d_ws, size_t ws_size,
                              hipStream_t stream) {
    const float* x    = (const float*)d_in[0];   // (1,8,361,720) fp32
    const float* pctw = (const float*)d_in[1];   // (361,360,361) fp32
    float* out = (float*)d_out;                  // (1,8,360,361) complex64 -> float pairs

    float* Wp = (float*)d_ws;                    // 720*736 floats (2.1 MB)
    float* Xt = Wp + (size_t)NLON * WCOLS;       // 361*181*16*2 floats (8.4 MB)

    // 0) DFT matrix (deterministic rebuild every call)
    {
        int total = NLON * WCOLS;
        build_dft_matrix<<<(total + 255) / 256, 256, 0, stream>>>(Wp);
    }
    // 1) rfft-as-GEMM on the fp32 matrix pipe
    dft_wmma<<<dim3(RTILES, (CTILES + 3) / 4), 128, 0, stream>>>(x, Wp, Xt);
    // 2) per-m Legendre contraction on the fp32 matrix pipe
    legendre_wmma<<<dim3(MMAX, (LTILES + 3) / 4), 128, 0, stream>>>(pctw, Xt, out);
}